// Model_53257594470855
// MI455X (gfx1250) — compile-verified
//
#include <hip/hip_runtime.h>

// ---------------------------------------------------------------------------
// GraphSAGE block conv + cross-partition merge for MI455X (gfx1250, wave32).
//
// P=4 partitions, N=65536 nodes, D=O=128, E=500000 edges/block, M=32768.
//
// Per (s,d) block:   h_neigh = segment_mean(feats[s][esrc], edst)
//                    out     = feats[s][:num_dst] @ Ws[s] + h_neigh @ Wn[s] + b[s]
// Merge:             final[d] = out[d][d]; final[d][midx[s,d]] += out[s][d]
//
// GEMM uses V_WMMA_F32_16X16X4_F32 (exact f32), one 16x128 row strip per
// wave (8 accumulators), W staged in LDS in 64-row pair-interleaved chunks.
// ---------------------------------------------------------------------------

#define PNUM 4
#define NNODE 65536
#define DDIM 128
#define ODIM 128
#define ENUM 500000
#define MNUM 32768

typedef __attribute__((ext_vector_type(2))) float v2f;
typedef __attribute__((ext_vector_type(8))) float v8f;

// ---------------------------------------------------------------------------
// Edge aggregation: agg[dst] += feats[src]; deg[dst] += 1.
// One thread per (edge, 4-float chunk): float4 gather + 4 f32 global atomics.
// Feature tables are L2-resident (32 MB/partition vs 192 MB L2).
// ---------------------------------------------------------------------------
__global__ __launch_bounds__(256) void sage_aggregate(
    const float* __restrict__ featsS,
    const int* __restrict__ esrc, const int* __restrict__ edst,
    float* __restrict__ agg, float* __restrict__ deg)
{
    int t = blockIdx.x * 256 + threadIdx.x;      // E*32 threads exactly
    int e = t >> 5;
    int chunk = t & 31;
    int src = esrc[e];
    int dst = edst[e];
    const float4 v = *(const float4*)(featsS + (size_t)src * DDIM + chunk * 4);
    float* a = agg + (size_t)dst * DDIM + chunk * 4;
    unsafeAtomicAdd(a + 0, v.x);
    unsafeAtomicAdd(a + 1, v.y);
    unsafeAtomicAdd(a + 2, v.z);
    unsafeAtomicAdd(a + 3, v.w);
    if (chunk == 0) unsafeAtomicAdd(deg + dst, 1.0f);
}

// ---------------------------------------------------------------------------
// Fused K=256 GEMM: out = [feat_dst | agg/max(deg,1)] @ [Ws ; Wn] + b.
// Block = 256 threads = 8 waves; wave w computes rows [16*(8*bid+w), +16).
// W staged in LDS 64 k-rows at a time, pair-interleaved so a B fragment
// (two k-adjacent elements of one column) is a single ds_load_b64.
//
// V_WMMA_F32_16X16X4_F32 operand layout (ISA 7.12.2):
//   A 16x4 : lane l<16 -> row l,  vgpr0/1 = K,K+1 ; lane l+16 -> K+2,K+3
//   B 4x16 : lane l<16 -> col l,  vgpr0/1 = K,K+1 ; lane l+16 -> K+2,K+3
//   C 16x16: lane l, vgpr v -> row v + 8*(l>=16), col l%16
// ---------------------------------------------------------------------------
__global__ __launch_bounds__(256) void sage_gemm(
    const float* __restrict__ featsS,
    const float* __restrict__ Ws, const float* __restrict__ Wn,
    const float* __restrict__ bias,
    const float* __restrict__ agg, const float* __restrict__ deg,
    const int* __restrict__ midx,       // nullptr => diagonal block (store)
    float* __restrict__ outD)           // final[d], [N,O]
{
    __shared__ float ldsW[64 * DDIM];   // 32 KB, pair-interleaved

    const int tid  = threadIdx.x;
    const int lane = tid & 31;
    const int wv   = tid >> 5;
    const int l16  = lane & 15;
    const int half = lane >> 4;         // 0: K,K+1   1: K+2,K+3
    const int koff = half * 2;

    const int r0   = (blockIdx.x * 8 + wv) * 16;
    const int aRow = r0 + l16;

    // accumulators init with bias (bias depends on column only)
    v8f c[8];
#pragma unroll
    for (int t = 0; t < 8; ++t) {
        float bv = bias[t * 16 + l16];
#pragma unroll
        for (int j = 0; j < 8; ++j) c[t][j] = bv;
    }

    const float rscale = 1.0f / fmaxf(deg[aRow], 1.0f);

    for (int ch = 0; ch < 4; ++ch) {
        // ---- stage 64 k-rows of W into LDS (pairs of k-adjacent rows) ----
        const float* Wblk = (ch < 2) ? Ws : Wn;
        const int kbase = (ch & 1) * 64;
#pragma unroll
        for (int i = tid; i < 32 * DDIM; i += 256) {   // 16 iters, no divergence
            int p   = i >> 7;                          // pair index 0..31
            int col = i & 127;
            int k   = kbase + p * 2;
            ldsW[i * 2 + 0] = Wblk[k * ODIM + col];
            ldsW[i * 2 + 1] = Wblk[(k + 1) * ODIM + col];
        }
        __syncthreads();

        const float* Asrc   = (ch < 2) ? featsS : agg;
        const float  ascale = (ch < 2) ? 1.0f : rscale;
        const float* aptr   = Asrc + (size_t)aRow * DDIM + kbase + koff;

#pragma unroll
        for (int ks = 0; ks < 64; ks += 4) {
            v2f a = *(const v2f*)(aptr + ks);
            a[0] *= ascale;
            a[1] *= ascale;
            const int p = (ks >> 1) + half;            // LDS pair for this half
            const float* lrow = &ldsW[p * DDIM * 2];
#pragma unroll
            for (int t = 0; t < 8; ++t) {
                v2f b = *(const v2f*)(lrow + (t * 16 + l16) * 2);
                c[t] = __builtin_amdgcn_wmma_f32_16x16x4_f32(
                    false, a, false, b, (short)0, c[t], false, false);
            }
        }
        __syncthreads();
    }

    // ---- writeback ----
    if (midx == nullptr) {
        // diagonal block: plain stores fully initialize final[d]
#pragma unroll
        for (int v = 0; v < 8; ++v) {
            int row = r0 + half * 8 + v;
#pragma unroll
            for (int t = 0; t < 8; ++t)
                outD[(size_t)row * ODIM + t * 16 + l16] = c[t][v];
        }
    } else {
        // off-diagonal: scatter-add through merge indices
        int mrow[8];
#pragma unroll
        for (int v = 0; v < 8; ++v) mrow[v] = midx[r0 + half * 8 + v];
#pragma unroll
        for (int v = 0; v < 8; ++v) {
#pragma unroll
            for (int t = 0; t < 8; ++t)
                unsafeAtomicAdd(&outD[(size_t)mrow[v] * ODIM + t * 16 + l16], c[t][v]);
        }
    }
}

// ---------------------------------------------------------------------------
// Driver: for each dest partition d, run the diagonal block first (stores),
// then the three off-diagonal blocks (atomic-add merges). One reusable
// agg/deg workspace (~34 MB), zeroed per block. All graph-capture safe.
// ---------------------------------------------------------------------------
extern "C" void kernel_launch(void* const* d_in, const int* in_sizes, int n_in,
                              void* d_out, int out_size, void* d_ws, size_t ws_size,
                              hipStream_t stream) {
    const float* feats  = (const float*)d_in[0];   // [P,N,D]
    const float* Wself  = (const float*)d_in[1];   // [P,D,O]
    const float* Wneigh = (const float*)d_in[2];   // [P,D,O]
    const float* bias   = (const float*)d_in[3];   // [P,O]
    const int*   esrc   = (const int*)d_in[4];     // [P,P,E]
    const int*   edst   = (const int*)d_in[5];     // [P,P,E]
    const int*   midx   = (const int*)d_in[6];     // [P,P,M]
    float* out = (float*)d_out;                    // [P,N,O]

    const size_t aggFloats = (size_t)NNODE * DDIM;
    float* agg = (float*)d_ws;
    float* deg = agg + aggFloats;
    const size_t clearBytes = (aggFloats + NNODE) * sizeof(float);

    const int aggBlocks = (ENUM * 32) / 256;       // 62500, exact

    for (int d = 0; d < PNUM; ++d) {
        for (int ss = 0; ss < PNUM; ++ss) {
            const int s = (ss + d) % PNUM;         // ss==0 -> diagonal first
            const int numDst = (s == d) ? NNODE : MNUM;
            const size_t blk = (size_t)s * PNUM + d;

            hipMemsetAsync(d_ws, 0, clearBytes, stream);

            sage_aggregate<<<aggBlocks, 256, 0, stream>>>(
                feats + (size_t)s * NNODE * DDIM,
                esrc + blk * ENUM, edst + blk * ENUM, agg, deg);

            sage_gemm<<<numDst / 128, 256, 0, stream>>>(
                feats + (size_t)s * NNODE * DDIM,
                Wself + (size_t)s * DDIM * ODIM,
                Wneigh + (size_t)s * DDIM * ODIM,
                bias + (size_t)s * ODIM,
                agg, deg,
                (s == d) ? nullptr : (midx + blk * MNUM),
                out + (size_t)d * NNODE * ODIM);
        }
    }
}